// NeusRender_85100482003215
// MI455X (gfx1250) — compile-verified
//
#include <hip/hip_runtime.h>
#include <stdint.h>

#define N_RAYS    65536
#define N_SAMPLES 128
#define BLOCK     256
#define CH        8                       // samples per chunk
#define NCHUNK    (N_SAMPLES / CH)        // 16

// TDM pads 1 dword after every 32 dwords stored -> conflict-free LDS rows
#define CHUNK_DW     (BLOCK * CH)                 // 2048
#define CHUNK_DW_PAD (CHUNK_DW + CHUNK_DW / 32)   // 2112

typedef unsigned int u32x4 __attribute__((ext_vector_type(4)));
typedef int          i32x8 __attribute__((ext_vector_type(8)));
typedef int          i32x4 __attribute__((ext_vector_type(4)));
typedef float        f32x4 __attribute__((ext_vector_type(4)));

// ---------------------------------------------------------------------------
// Tensor Data Mover: 2-D tile load (tile_w elements x tile_h rows) from a
// row-major tensor with row stride `row_stride` (elements, 4B each) into LDS
// at byte address lds_addr. D# packed per CDNA5 ISA.
// pad_enable=1, pad_interval=32 dwords (code 4), pad_amount=1 dword (code 0).
// 6-arg builtin form (amdgpu-toolchain / clang-23).
// ---------------------------------------------------------------------------
__device__ __forceinline__ void tdm_load_tile(uint32_t lds_addr,
                                              const float* gsrc,
                                              uint32_t tile_w, uint32_t tile_h,
                                              uint32_t row_stride) {
  const uint64_t ga = (uint64_t)(uintptr_t)gsrc;

  u32x4 g0;
  g0[0] = 1u;                                   // count=1 valid descriptor
  g0[1] = lds_addr;                             // lds_addr [63:32]
  g0[2] = (uint32_t)(ga & 0xFFFFFFFFu);         // global_addr[31:0]
  g0[3] = (uint32_t)((ga >> 32) & 0x1FFFFFFu)   // global_addr[56:32]
        | (2u << 30);                           // type=2 ("image")

  const uint32_t TDIM = 0x00FFFFFFu;            // huge tensor dims: no OOB clip
  i32x8 g1;
  g1[0] = (int)((2u << 16)      // data_size = 4 bytes
              | (1u << 20)      // pad_enable
              | (4u << 22)      // pad_interval: 32 dwords
              | (0u << 25));    // pad_amount: 1 dword
  g1[1] = (int)((TDIM & 0xFFFFu) << 16);                              // tdim0 lo
  g1[2] = (int)(((TDIM >> 16) & 0xFFFFu) | ((TDIM & 0xFFFFu) << 16)); // tdim0 hi | tdim1 lo
  g1[3] = (int)(((TDIM >> 16) & 0xFFFFu) | (tile_w << 16));           // tdim1 hi | tile_dim0
  g1[4] = (int)(tile_h & 0xFFFFu);                                    // tile_dim1, tile_dim2=0
  g1[5] = (int)row_stride;                                            // tensor_dim0_stride[31:0]
  g1[6] = 0;
  g1[7] = 0;

  i32x4 g2;                       // dims 2/3 large, strides 0 (unused, z==0)
  g2[0] = (int)0x00FFFFFF;        // tensor_dim2
  g2[1] = (int)0x00FFFFFF;        // tensor_dim3
  g2[2] = 0;
  g2[3] = 0;                      // tile_dim3 = 0

  i32x4 g3;
  g3[0] = 0;                      // tensor_dim3_stride
  g3[1] = (int)0xFFFF0000u;       // tensor_dim4 lo in [31:16]
  g3[2] = (int)0x0000FFFFu;       // tensor_dim4 hi, tile_dim4 = 0
  g3[3] = 0;

  i32x8 g4 = {0, 0, 0, 0, 0, 0, 0, 0};  // extra group in 6-arg form (unused)

  __builtin_amdgcn_tensor_load_to_lds(g0, g1, g2, g3, g4, /*cpol=*/0);
}

__global__ __launch_bounds__(BLOCK) void neus_render_kernel(
    const float* __restrict__ sdf,    // [R,S]
    const float* __restrict__ color,  // [R,S,3]
    const float* __restrict__ zv,     // [R,S]
    const float* __restrict__ sp,     // [1]
    const float* __restrict__ bg,     // [3]
    float* __restrict__ out)          // pixel[3R] | invdepth[R] | weight[R*S]
{
  __shared__ float sdf_s[2][CHUNK_DW_PAD];
  __shared__ float z_s  [2][CHUNK_DW_PAD];

  const int t  = threadIdx.x;
  const int rg = blockIdx.x * BLOCK + t;
  const float sv = sp[0];
  const uint64_t row0 = (uint64_t)blockIdx.x * BLOCK;   // first ray of block

  // Prime the pipeline: chunk 0 tiles for sdf and z (wave 0 only; TENSORcnt
  // is per-wave, other waves sync through s_barrier below).
  if (t < 32) {
    tdm_load_tile((uint32_t)(uintptr_t)&sdf_s[0][0],
                  sdf + row0 * N_SAMPLES, CH, BLOCK, N_SAMPLES);
    tdm_load_tile((uint32_t)(uintptr_t)&z_s[0][0],
                  zv  + row0 * N_SAMPLES, CH, BLOCK, N_SAMPLES);
  }

  float P = 1.0f;          // running prod (1 - alpha[j])  (unclipped, faithful)
  float sig_prev = 0.0f;
  float wsum = 0.0f, invd = 0.0f, pr = 0.0f, pg = 0.0f, pb = 0.0f;
  float z_prev = 1.0f, cpr = 0.0f, cpg = 0.0f, cpb = 0.0f;
  f32x4 wq = {0.0f, 0.0f, 0.0f, 0.0f};

  // conflict-free padded row base for this thread: 8t + floor(8t/32)
  const int lidx = t * CH + (t >> 2);
  float* __restrict__ wout = out + 4 * (size_t)N_RAYS + (size_t)rg * N_SAMPLES;

  for (int c = 0; c < NCHUNK; ++c) {
    const int buf = c & 1;
    if (t < 32) {
      if (c + 1 < NCHUNK) {   // prefetch next chunk into other buffer
        tdm_load_tile((uint32_t)(uintptr_t)&sdf_s[buf ^ 1][0],
                      sdf + row0 * N_SAMPLES + (uint64_t)(c + 1) * CH,
                      CH, BLOCK, N_SAMPLES);
        tdm_load_tile((uint32_t)(uintptr_t)&z_s[buf ^ 1][0],
                      zv  + row0 * N_SAMPLES + (uint64_t)(c + 1) * CH,
                      CH, BLOCK, N_SAMPLES);
        __builtin_amdgcn_s_wait_tensorcnt(2);  // current chunk's 2 ops done
      } else {
        __builtin_amdgcn_s_wait_tensorcnt(0);
      }
    }
    __syncthreads();   // LDS tiles for chunk c are ready for all waves

    // color chunk: 24 contiguous floats per thread, 16B-aligned float4 loads
    float cc[CH * 3];
    {
      const f32x4* c4 = (const f32x4*)(color + (size_t)rg * (N_SAMPLES * 3)
                                             + (size_t)c * (CH * 3));
#pragma unroll
      for (int q = 0; q < (CH * 3) / 4; ++q) {
        f32x4 v = c4[q];
        cc[q * 4 + 0] = v.x; cc[q * 4 + 1] = v.y;
        cc[q * 4 + 2] = v.z; cc[q * 4 + 3] = v.w;
      }
    }

#pragma unroll
    for (int x = 0; x < CH; ++x) {
      const float sdv = sdf_s[buf][lidx + x];
      const float sig = 1.0f / (1.0f + expf(-sv * sdv));
      if (x > 0 || c > 0) {                 // i = c*CH + x > 0
        const int j = c * CH + x - 1;       // emit weight[j]
        const float a  = (sig_prev - sig) / sig_prev;   // alpha[j] (unclipped)
        const float ac = fminf(fmaxf(a, 0.0f), 1.0f);
        const float w  = (j == 0) ? 0.0f : P * ac;      // trans[0] == 0
        P *= (1.0f - a);

        const float zj  = (x == 0) ? z_prev : z_s[buf][lidx + x - 1];
        const float crj = (x == 0) ? cpr : cc[(x - 1) * 3 + 0];
        const float cgj = (x == 0) ? cpg : cc[(x - 1) * 3 + 1];
        const float cbj = (x == 0) ? cpb : cc[(x - 1) * 3 + 2];

        wsum += w;
        invd += w / zj;
        pr = fmaf(w, crj, pr);
        pg = fmaf(w, cgj, pg);
        pb = fmaf(w, cbj, pb);

        wq[(x - 1) & 3] = w;                // (j & 3) constant after unroll
        if (((x - 1) & 3) == 3)
          __builtin_nontemporal_store(wq, (f32x4*)(wout + (j - 3)));
      }
      sig_prev = sig;
    }
    // carry last sample of this chunk across the buffer swap
    z_prev = z_s[buf][lidx + CH - 1];
    cpr = cc[(CH - 1) * 3 + 0];
    cpg = cc[(CH - 1) * 3 + 1];
    cpb = cc[(CH - 1) * 3 + 2];
    __syncthreads();   // everyone done reading buf before next TDM overwrites
  }

  // weight[127] = trans * alpha_appended(=0) = 0; flush last float4 (j=124..127)
  wq[3] = 0.0f;
  __builtin_nontemporal_store(wq, (f32x4*)(wout + (N_SAMPLES - 4)));

  const float resid = 1.0f - wsum;
  out[(size_t)rg * 3 + 0] = fmaf(resid, bg[0], pr);
  out[(size_t)rg * 3 + 1] = fmaf(resid, bg[1], pg);
  out[(size_t)rg * 3 + 2] = fmaf(resid, bg[2], pb);
  out[3 * (size_t)N_RAYS + rg] = invd;
}

extern "C" void kernel_launch(void* const* d_in, const int* in_sizes, int n_in,
                              void* d_out, int out_size, void* d_ws, size_t ws_size,
                              hipStream_t stream) {
  const float* sdf   = (const float*)d_in[0];
  const float* color = (const float*)d_in[1];
  const float* zv    = (const float*)d_in[2];
  const float* s     = (const float*)d_in[3];
  const float* bg    = (const float*)d_in[4];
  float* out = (float*)d_out;

  neus_render_kernel<<<N_RAYS / BLOCK, BLOCK, 0, stream>>>(sdf, color, zv, s, bg, out);
}